// DecoderWithAttention_25632364822546
// MI455X (gfx1250) — compile-verified
//
#include <hip/hip_runtime.h>
#include <math.h>

// Problem constants (match reference)
#define B_SZ  16
#define T_DEC 32
#define F_DIM 8
#define S_ENC 256
#define H_DIM 512
#define XK    (F_DIM + H_DIM)     // 520 = GRU0 input width
#define OUTK  (2 * H_DIM + F_DIM) // 1032

typedef __attribute__((ext_vector_type(2))) float v2f;
typedef __attribute__((ext_vector_type(8))) float v8f;

static __device__ __forceinline__ v8f vzero8() {
  v8f v;
#pragma unroll
  for (int i = 0; i < 8; ++i) v[i] = 0.0f;
  return v;
}

// D = A(16x4) * B(4x16) + C, full fp32 (V_WMMA_F32_16X16X4_F32)
static __device__ __forceinline__ v8f wmma4(v2f a, v2f b, v8f c) {
  return __builtin_amdgcn_wmma_f32_16x16x4_f32(false, a, false, b, (short)0, c,
                                               false, false);
}

static __device__ __forceinline__ float sigmoidf_(float x) {
  return 1.0f / (1.0f + expf(-x));
}

// ---------------------------------------------------------------------------
// init: h0 = hidden0[0], h1 = hidden0[1], x_in = target[:,0,:]
// ---------------------------------------------------------------------------
__global__ __launch_bounds__(256) void init_state(
    const float* __restrict__ hidden0, const float* __restrict__ target,
    float* __restrict__ h0, float* __restrict__ h1, float* __restrict__ x_in) {
  int i = blockIdx.x * blockDim.x + threadIdx.x;
  if (i < B_SZ * H_DIM) {
    h0[i] = hidden0[i];
    h1[i] = hidden0[B_SZ * H_DIM + i];
  }
  if (i < B_SZ * F_DIM) {
    int b = i >> 3, f = i & 7;
    x_in[i] = target[(size_t)b * T_DEC * F_DIM + f]; // target[b, 0, f]
  }
}

// ---------------------------------------------------------------------------
// Time-invariant precompute: encW2[m, j] = sum_k enc[m,k]*attn_W[j,H+k] + attn_b[j]
// M = B*S = 4096, N = H = 512, K = 512.  One wave per 16x16 tile.
// ---------------------------------------------------------------------------
__global__ __launch_bounds__(128) void enc_precompute(
    const float* __restrict__ enc,     // [B*S, H]
    const float* __restrict__ attn_W,  // [H, 2H]
    const float* __restrict__ attn_b,  // [H]
    float* __restrict__ encW2) {       // [B*S, H]
  int wave = (blockIdx.x * blockDim.x + threadIdx.x) >> 5; // 0..8191
  int lane = threadIdx.x & 31;
  int mt = wave >> 5;
  int nt = wave & 31;
  int m = mt * 16 + (lane & 15);
  int n = nt * 16 + (lane & 15);
  int kg = (lane >> 4) * 2; // lane-group K offset (ISA A/B layout)
  const float* arow = enc + (size_t)m * H_DIM;
  const float* brow = attn_W + (size_t)n * (2 * H_DIM) + H_DIM; // B[k][n]=W[n][H+k]
  v8f acc = vzero8();
  for (int kk = 0; kk < H_DIM; kk += 4) {
    v2f a = *(const v2f*)(arow + kk + kg);
    v2f b = *(const v2f*)(brow + kk + kg);
    acc = wmma4(a, b, acc);
  }
  float bias = attn_b[n];
  int mbase = mt * 16 + (lane >> 4) * 8;
  float* dcol = encW2 + (size_t)mbase * H_DIM + n;
#pragma unroll
  for (int v = 0; v < 8; ++v) dcol[(size_t)v * H_DIM] = acc[v] + bias;
}

// ---------------------------------------------------------------------------
// Fused per-step kernel, one block per batch row (16 x 256):
//   Phase 0: output projection of step t-1 (relu), becomes x_in(t); writes d_out
//   Phase 1: hW1[b,:] = h_top[b,:] @ W1.T via WMMA (full 16-row tile, keep row b)
//   Phase 2: scores = v . tanh(hW1 + encW2[b,s]); softmax over s
//   Phase 3: weighted[b,:] = sum_s w[s]*enc[b,s,:]; assemble x_cat
// ---------------------------------------------------------------------------
__global__ __launch_bounds__(S_ENC) void attention_step(
    const float* __restrict__ h_top,   // [16,H]: h1 entering step t == h1_new(t-1)
    const float* __restrict__ encW2,   // [B*S, H]
    const float* __restrict__ enc,     // [B, S, H]
    const float* __restrict__ v_w,     // [H]
    const float* __restrict__ attn_W,  // [H, 2H]
    const float* __restrict__ out_W,   // [F, OUTK]
    const float* __restrict__ out_b,   // [F]
    const float* __restrict__ w_prev,  // weighted(t-1)
    float* __restrict__ w_cur,         // weighted(t)
    const float* __restrict__ x_prev,  // x_in(t-1)  (t==0: x_in(0))
    float* __restrict__ x_cur,         // x_in(t) store
    float* __restrict__ x_cat,         // [16, XK]
    float* __restrict__ d_out, int t) {
  __shared__ float sh_hw[H_DIM];
  __shared__ float sh_vw[H_DIM];
  __shared__ float red[S_ENC];
  __shared__ float shw[S_ENC];
  __shared__ float sh_out[F_DIM];
  int b = blockIdx.x;
  int tid = threadIdx.x;
  int wave = tid >> 5, lane = tid & 31;

  // stage v_w in LDS (used 512x per thread in phase 2)
  for (int j = tid; j < H_DIM; j += S_ENC) sh_vw[j] = v_w[j];

  // ---- Phase 0: deferred output projection of step t-1 ----
  if (t > 0) {
    int f = wave; // 8 waves <-> 8 features
    const float* w = out_W + (size_t)f * OUTK;
    float p = 0.0f;
    for (int j = lane; j < H_DIM; j += 32) p += h_top[b * H_DIM + j] * w[j];
    for (int j = lane; j < H_DIM; j += 32)
      p += w_prev[b * H_DIM + j] * w[H_DIM + j];
    if (lane < F_DIM) p += x_prev[b * F_DIM + lane] * w[2 * H_DIM + lane];
#pragma unroll
    for (int off = 16; off > 0; off >>= 1) p += __shfl_down(p, off);
    if (lane == 0) {
      float o = fmaxf(p + out_b[f], 0.0f);
      sh_out[f] = o;
      x_cur[b * F_DIM + f] = o;                           // x_in(t) = out(t-1)
      d_out[((size_t)b * T_DEC + (t - 1)) * F_DIM + f] = o;
    }
  } else {
    if (tid < F_DIM) {
      float o = x_prev[b * F_DIM + tid]; // initial x_in from target[:,0,:]
      sh_out[tid] = o;
      x_cur[b * F_DIM + tid] = o;
    }
  }
  __syncthreads();
  if (tid < F_DIM) x_cat[b * XK + tid] = sh_out[tid];

  // ---- Phase 1: hW1 via WMMA (each wave: 4 of 32 N-tiles, K=512) ----
  {
    int kg = (lane >> 4) * 2;
    const float* arow = h_top + (size_t)(lane & 15) * H_DIM;
    for (int tt = wave; tt < 32; tt += 8) {
      int n = tt * 16 + (lane & 15);
      const float* brow = attn_W + (size_t)n * (2 * H_DIM); // B[k][n] = W[n][k]
      v8f acc = vzero8();
      for (int kk = 0; kk < H_DIM; kk += 4) {
        v2f a = *(const v2f*)(arow + kk + kg);
        v2f bb = *(const v2f*)(brow + kk + kg);
        acc = wmma4(a, bb, acc);
      }
      // keep row m == b: vgpr index b&7, lane half b>>3
      if ((lane >> 4) == ((b >> 3) & 1))
        sh_hw[tt * 16 + (lane & 15)] = acc[b & 7];
    }
  }
  __syncthreads();

  // ---- Phase 2: scores + softmax over s ----
  int s = tid;
  const float* e = encW2 + ((size_t)b * S_ENC + s) * H_DIM;
  float sc = 0.0f;
  for (int j = 0; j < H_DIM; ++j) sc += sh_vw[j] * tanhf(sh_hw[j] + e[j]);

  red[s] = sc;
  __syncthreads();
  for (int off = S_ENC / 2; off > 0; off >>= 1) {
    if (s < off) red[s] = fmaxf(red[s], red[s + off]);
    __syncthreads();
  }
  float mx = red[0];
  __syncthreads();
  float ex = expf(sc - mx);
  red[s] = ex;
  __syncthreads();
  for (int off = S_ENC / 2; off > 0; off >>= 1) {
    if (s < off) red[s] += red[s + off];
    __syncthreads();
  }
  shw[s] = ex / red[0];
  __syncthreads();

  // ---- Phase 3: weighted sum (coalesced over h) ----
  for (int h = s; h < H_DIM; h += S_ENC) {
    const float* ebase = enc + (size_t)b * S_ENC * H_DIM + h;
    float acc = 0.0f;
    for (int s2 = 0; s2 < S_ENC; ++s2)
      acc += shw[s2] * ebase[(size_t)s2 * H_DIM];
    w_cur[b * H_DIM + h] = acc;
    x_cat[b * XK + F_DIM + h] = acc;
  }
}

// ---------------------------------------------------------------------------
// One GRU layer step: each wave owns a 16-wide hidden-unit tile (32 waves),
// six WMMA K-chains (r/z/n vs Wih and Whh), fused gate math.
// PyTorch gate order: rows [0,H)=r, [H,2H)=z, [2H,3H)=n.
// ---------------------------------------------------------------------------
__global__ __launch_bounds__(128) void gru_layer(
    const float* __restrict__ x,      // [16, kdim]
    int kdim,                         // 520 (layer0) or 512 (layer1)
    const float* __restrict__ h_prev, // [16, H]
    const float* __restrict__ Wih,    // [3H, kdim]
    const float* __restrict__ Whh,    // [3H, H]
    const float* __restrict__ bih, const float* __restrict__ bhh,
    float* __restrict__ h_new) {      // [16, H]
  int wave = (blockIdx.x * blockDim.x + threadIdx.x) >> 5; // 0..31
  int lane = threadIdx.x & 31;
  int n = wave * 16 + (lane & 15);
  int kg = (lane >> 4) * 2;

  const float* ax = x + (size_t)(lane & 15) * kdim;
  const float* ah = h_prev + (size_t)(lane & 15) * H_DIM;
  const float* bri = Wih + (size_t)n * kdim;
  const float* bzi = Wih + (size_t)(H_DIM + n) * kdim;
  const float* bni = Wih + (size_t)(2 * H_DIM + n) * kdim;
  const float* brh = Whh + (size_t)n * H_DIM;
  const float* bzh = Whh + (size_t)(H_DIM + n) * H_DIM;
  const float* bnh = Whh + (size_t)(2 * H_DIM + n) * H_DIM;

  v8f cri = vzero8(), czi = vzero8(), cni = vzero8();
  v8f crh = vzero8(), czh = vzero8(), cnh = vzero8();

  for (int kk = 0; kk < kdim; kk += 4) {
    v2f a = *(const v2f*)(ax + kk + kg);
    cri = wmma4(a, *(const v2f*)(bri + kk + kg), cri);
    czi = wmma4(a, *(const v2f*)(bzi + kk + kg), czi);
    cni = wmma4(a, *(const v2f*)(bni + kk + kg), cni);
  }
  for (int kk = 0; kk < H_DIM; kk += 4) {
    v2f a = *(const v2f*)(ah + kk + kg);
    crh = wmma4(a, *(const v2f*)(brh + kk + kg), crh);
    czh = wmma4(a, *(const v2f*)(bzh + kk + kg), czh);
    cnh = wmma4(a, *(const v2f*)(bnh + kk + kg), cnh);
  }

  float bir = bih[n], biz = bih[H_DIM + n], bin = bih[2 * H_DIM + n];
  float bhr = bhh[n], bhz = bhh[H_DIM + n], bhn = bhh[2 * H_DIM + n];
  int mbase = (lane >> 4) * 8;
#pragma unroll
  for (int v = 0; v < 8; ++v) {
    int m = mbase + v;
    float hp = h_prev[(size_t)m * H_DIM + n];
    float r = sigmoidf_(cri[v] + bir + crh[v] + bhr);
    float z = sigmoidf_(czi[v] + biz + czh[v] + bhz);
    float nn = tanhf(cni[v] + bin + r * (cnh[v] + bhn));
    h_new[(size_t)m * H_DIM + n] = (1.0f - z) * nn + z * hp;
  }
}

// ---------------------------------------------------------------------------
// Standalone output projection (only used for the final step t=31)
// ---------------------------------------------------------------------------
__global__ __launch_bounds__(128) void out_proj(
    const float* __restrict__ h1, const float* __restrict__ weighted,
    const float* __restrict__ x_in_cur, const float* __restrict__ out_W,
    const float* __restrict__ out_b, float* __restrict__ d_out, int t,
    float* __restrict__ x_in_next) {
  int tid = threadIdx.x; // 128 = 16 batches * 8 features
  int b = tid >> 3, f = tid & 7;
  const float* w = out_W + (size_t)f * OUTK;
  const float* h = h1 + (size_t)b * H_DIM;
  const float* wt = weighted + (size_t)b * H_DIM;
  float acc = out_b[f];
  for (int j = 0; j < H_DIM; ++j) acc += h[j] * w[j];
  for (int j = 0; j < H_DIM; ++j) acc += wt[j] * w[H_DIM + j];
  for (int j = 0; j < F_DIM; ++j)
    acc += x_in_cur[b * F_DIM + j] * w[2 * H_DIM + j];
  acc = fmaxf(acc, 0.0f);
  d_out[((size_t)b * T_DEC + t) * F_DIM + f] = acc;
  x_in_next[b * F_DIM + f] = acc;
}

// ---------------------------------------------------------------------------
extern "C" void kernel_launch(void* const* d_in, const int* in_sizes, int n_in,
                              void* d_out, int out_size, void* d_ws,
                              size_t ws_size, hipStream_t stream) {
  const float* target = (const float*)d_in[0];
  const float* hidden0 = (const float*)d_in[1];
  const float* enc = (const float*)d_in[2];
  const float* attn_W = (const float*)d_in[3];
  const float* attn_b = (const float*)d_in[4];
  const float* v_w = (const float*)d_in[5];
  const float* Wih0 = (const float*)d_in[6];
  const float* Whh0 = (const float*)d_in[7];
  const float* bih0 = (const float*)d_in[8];
  const float* bhh0 = (const float*)d_in[9];
  const float* Wih1 = (const float*)d_in[10];
  const float* Whh1 = (const float*)d_in[11];
  const float* bih1 = (const float*)d_in[12];
  const float* bhh1 = (const float*)d_in[13];
  const float* out_W = (const float*)d_in[14];
  const float* out_b = (const float*)d_in[15];
  float* out = (float*)d_out;

  float* ws = (float*)d_ws;
  float* encW2 = ws;                            ws += (size_t)B_SZ * S_ENC * H_DIM;
  float* x_cat = ws;                            ws += B_SZ * XK;
  float* wbuf[2] = {ws, ws + B_SZ * H_DIM};     ws += 2 * B_SZ * H_DIM;
  float* h0buf[2] = {ws, ws + B_SZ * H_DIM};    ws += 2 * B_SZ * H_DIM;
  float* h1buf[2] = {ws, ws + B_SZ * H_DIM};    ws += 2 * B_SZ * H_DIM;
  float* xibuf[2] = {ws, ws + B_SZ * F_DIM};    ws += 2 * B_SZ * F_DIM;

  init_state<<<32, 256, 0, stream>>>(hidden0, target, h0buf[0], h1buf[0],
                                     xibuf[0]);
  enc_precompute<<<2048, 128, 0, stream>>>(enc, attn_W, attn_b, encW2);

  for (int t = 0; t < T_DEC; ++t) {
    int p = t & 1, q = (t + 1) & 1;
    const float* xprev = (t == 0) ? xibuf[0] : xibuf[q]; // x_in(t-1)
    attention_step<<<B_SZ, S_ENC, 0, stream>>>(
        h1buf[p], encW2, enc, v_w, attn_W, out_W, out_b,
        /*w_prev=*/wbuf[q], /*w_cur=*/wbuf[p], xprev, /*x_cur=*/xibuf[p],
        x_cat, out, t);
    gru_layer<<<8, 128, 0, stream>>>(x_cat, XK, h0buf[p], Wih0, Whh0, bih0,
                                     bhh0, h0buf[q]);
    gru_layer<<<8, 128, 0, stream>>>(h0buf[q], H_DIM, h1buf[p], Wih1, Whh1,
                                     bih1, bhh1, h1buf[q]);
  }
  // final output row (t = 31): h1_new(31)=h1buf[0], weighted(31)=wbuf[1],
  // x_in(31)=xibuf[1]
  out_proj<<<1, 128, 0, stream>>>(h1buf[0], wbuf[1], xibuf[1], out_W, out_b,
                                  out, 31, xibuf[0]);
}